// GraphSAGE_16295105921228
// MI455X (gfx1250) — compile-verified
//
#include <hip/hip_runtime.h>
#include <hip/hip_bf16.h>

typedef __attribute__((ext_vector_type(2))) float v2f;
typedef __attribute__((ext_vector_type(8))) float v8f;

// ---------------------------------------------------------------------------
// Kernel 0: zero out sums1[N*32], counts[N], sums2[N] (contiguous region)
// ---------------------------------------------------------------------------
__global__ void zero_kernel(float* __restrict__ p, int n) {
    int i = blockIdx.x * blockDim.x + threadIdx.x;
    if (i < n) p[i] = 0.0f;
}

// ---------------------------------------------------------------------------
// Kernel 1: counts[dst] += 1 per edge
// ---------------------------------------------------------------------------
__global__ void count_kernel(const int* __restrict__ ei, float* __restrict__ counts, int E) {
    int e = blockIdx.x * blockDim.x + threadIdx.x;
    if (e >= E) return;
    int dst = ei[E + e];
    atomicAdd(&counts[dst], 1.0f);
}

// ---------------------------------------------------------------------------
// Kernel 2: y_l = x @ w1_l, y_r = x @ w1_r using V_WMMA_F32_16X16X4_F32.
// One wave (32 lanes) computes a 16-row tile; 4 accumulators cover
// {w1_l,w1_r} x {cols 0-15, cols 16-31}. K-loop: 64 in steps of 4.
// A layout (ISA 7.12.2, 32-bit A 16x4): lanes 0-15 hold K=0,1 in v[0],v[1];
// lanes 16-31 hold K=2,3. B mirrored: lane half selects K pair, n = lane&15.
// ---------------------------------------------------------------------------
__global__ __launch_bounds__(256) void gemm_xw_kernel(
    const float* __restrict__ x,    // [N,64]
    const float* __restrict__ wl,   // [64,32]
    const float* __restrict__ wr,   // [64,32]
    float* __restrict__ yl,         // [N,32]
    float* __restrict__ yr,         // [N,32]
    int n_nodes, int n_tiles)
{
    int wave = (blockIdx.x * blockDim.x + threadIdx.x) >> 5;
    if (wave >= n_tiles) return;
    int lane = threadIdx.x & 31;
    int half = lane >> 4;          // 0: K={0,1}, 1: K={2,3} within each K4 step
    int l15  = lane & 15;
    int m0   = wave * 16;
    int arow = m0 + l15;
    int arow_c = arow < n_nodes ? arow : (n_nodes - 1);   // clamp for tail tile

    v8f acc_l0 = {}; v8f acc_l1 = {}; v8f acc_r0 = {}; v8f acc_r1 = {};

    const float* xrow = x + (long long)arow_c * 64 + 2 * half;
#pragma unroll
    for (int k = 0; k < 64; k += 4) {
        v2f a = *(const v2f*)(xrow + k);
        int kb = k + 2 * half;
        v2f bl0, bl1, br0, br1;
        bl0[0] = wl[kb * 32 + l15];        bl0[1] = wl[(kb + 1) * 32 + l15];
        bl1[0] = wl[kb * 32 + 16 + l15];   bl1[1] = wl[(kb + 1) * 32 + 16 + l15];
        br0[0] = wr[kb * 32 + l15];        br0[1] = wr[(kb + 1) * 32 + l15];
        br1[0] = wr[kb * 32 + 16 + l15];   br1[1] = wr[(kb + 1) * 32 + 16 + l15];
        acc_l0 = __builtin_amdgcn_wmma_f32_16x16x4_f32(false, a, false, bl0, (short)0, acc_l0, false, false);
        acc_l1 = __builtin_amdgcn_wmma_f32_16x16x4_f32(false, a, false, bl1, (short)0, acc_l1, false, false);
        acc_r0 = __builtin_amdgcn_wmma_f32_16x16x4_f32(false, a, false, br0, (short)0, acc_r0, false, false);
        acc_r1 = __builtin_amdgcn_wmma_f32_16x16x4_f32(false, a, false, br1, (short)0, acc_r1, false, false);
    }

    // D layout (32-bit C/D 16x16): lane half selects M offset 0/8; vgpr v -> M = v + 8*half
#pragma unroll
    for (int v = 0; v < 8; ++v) {
        int row = m0 + v + 8 * half;
        if (row < n_nodes) {
            long long o = (long long)row * 32;
            yl[o + l15]      = acc_l0[v];
            yl[o + 16 + l15] = acc_l1[v];
            yr[o + l15]      = acc_r0[v];
            yr[o + 16 + l15] = acc_r1[v];
        }
    }
}

// ---------------------------------------------------------------------------
// Kernel 3: sums1[dst,:] += y_l[src,:]  (8 threads per edge, float4 each)
// ---------------------------------------------------------------------------
__global__ void scatter32_kernel(const int* __restrict__ ei,
                                 const float* __restrict__ yl,
                                 float* __restrict__ sums1, int E) {
    long long t = (long long)blockIdx.x * blockDim.x + threadIdx.x;
    int e = (int)(t >> 3);
    if (e >= E) return;
    int c = ((int)t & 7) * 4;
    int src = ei[e];
    int dst = ei[E + e];
    const float4 v = *(const float4*)(yl + (long long)src * 32 + c);
    float* p = sums1 + (long long)dst * 32 + c;
    atomicAdd(p + 0, v.x);
    atomicAdd(p + 1, v.y);
    atomicAdd(p + 2, v.z);
    atomicAdd(p + 3, v.w);
}

// ---------------------------------------------------------------------------
// Kernel 4: per node: h = relu(sums1/cnt + y_r + b1)  (in registers)
//           z_l = h . w2_l ;  zr = h . w2_r + b2
// ---------------------------------------------------------------------------
__global__ void node_layer2_kernel(const float* __restrict__ sums1,
                                   const float* __restrict__ counts,
                                   const float* __restrict__ yr,
                                   const float* __restrict__ b1,
                                   const float* __restrict__ w2l,
                                   const float* __restrict__ w2r,
                                   const float* __restrict__ b2,
                                   float* __restrict__ zl,
                                   float* __restrict__ zr,
                                   int n) {
    int i = blockIdx.x * blockDim.x + threadIdx.x;
    if (i >= n) return;
    float inv = 1.0f / fmaxf(counts[i], 1.0f);
    long long o = (long long)i * 32;
    float dl = 0.0f, dr = 0.0f;
#pragma unroll
    for (int f = 0; f < 32; ++f) {
        float h = fmaf(sums1[o + f], inv, yr[o + f] + b1[f]);
        h = fmaxf(h, 0.0f);
        dl = fmaf(h, w2l[f], dl);
        dr = fmaf(h, w2r[f], dr);
    }
    zl[i] = dl;
    zr[i] = dr + b2[0];
}

// ---------------------------------------------------------------------------
// Kernel 5: sums2[dst] += z_l[src]  (one scalar atomic per edge)
// ---------------------------------------------------------------------------
__global__ void scatter1_kernel(const int* __restrict__ ei,
                                const float* __restrict__ zl,
                                float* __restrict__ sums2, int E) {
    int e = blockIdx.x * blockDim.x + threadIdx.x;
    if (e >= E) return;
    int src = ei[e];
    int dst = ei[E + e];
    atomicAdd(&sums2[dst], zl[src]);
}

// ---------------------------------------------------------------------------
// Kernel 6: out = sums2 / max(counts,1) + zr
// ---------------------------------------------------------------------------
__global__ void final_kernel(const float* __restrict__ sums2,
                             const float* __restrict__ counts,
                             const float* __restrict__ zr,
                             float* __restrict__ out, int n) {
    int i = blockIdx.x * blockDim.x + threadIdx.x;
    if (i >= n) return;
    out[i] = sums2[i] / fmaxf(counts[i], 1.0f) + zr[i];
}

// ---------------------------------------------------------------------------
extern "C" void kernel_launch(void* const* d_in, const int* in_sizes, int n_in,
                              void* d_out, int out_size, void* d_ws, size_t ws_size,
                              hipStream_t stream) {
    const float* x    = (const float*)d_in[0];   // [N,64]
    const int*   ei   = (const int*)  d_in[1];   // [2,E]
    const float* w1_l = (const float*)d_in[2];   // [64,32]
    const float* w1_r = (const float*)d_in[3];   // [64,32]
    const float* b1   = (const float*)d_in[4];   // [32]
    const float* w2_l = (const float*)d_in[5];   // [32,1]
    const float* w2_r = (const float*)d_in[6];   // [32,1]
    const float* b2   = (const float*)d_in[7];   // [1]

    const int N = in_sizes[0] / 64;
    const int E = in_sizes[1] / 2;

    // Workspace layout (floats), all naturally 16B-aligned for N multiple of 4:
    float* ws     = (float*)d_ws;
    float* sums1  = ws;                        // N*32
    float* counts = ws + (long long)N * 32;    // N
    float* sums2  = ws + (long long)N * 33;    // N
    float* yl     = ws + (long long)N * 34;    // N*32
    float* yr     = ws + (long long)N * 66;    // N*32
    float* zl     = ws + (long long)N * 98;    // N
    float* zr     = ws + (long long)N * 99;    // N
    float* outf   = (float*)d_out;

    const int B = 256;

    // 0) zero sums1 + counts + sums2 (contiguous N*34 floats)
    {
        int n = N * 34;
        zero_kernel<<<(n + B - 1) / B, B, 0, stream>>>(ws, n);
    }
    // 1) in-degree counts
    count_kernel<<<(E + B - 1) / B, B, 0, stream>>>(ei, counts, E);
    // 2) WMMA projections y_l = x@w1_l, y_r = x@w1_r
    {
        int n_tiles = (N + 15) / 16;
        int waves_per_block = B / 32;
        int blocks = (n_tiles + waves_per_block - 1) / waves_per_block;
        gemm_xw_kernel<<<blocks, B, 0, stream>>>(x, w1_l, w1_r, yl, yr, N, n_tiles);
    }
    // 3) edge scatter: sums1[dst] += y_l[src]
    {
        long long threads = (long long)E * 8;
        int blocks = (int)((threads + B - 1) / B);
        scatter32_kernel<<<blocks, B, 0, stream>>>(ei, yl, sums1, E);
    }
    // 4) node-local layer-2 projections
    node_layer2_kernel<<<(N + B - 1) / B, B, 0, stream>>>(
        sums1, counts, yr, b1, w2_l, w2_r, b2, zl, zr, N);
    // 5) edge scatter: sums2[dst] += z_l[src]
    scatter1_kernel<<<(E + B - 1) / B, B, 0, stream>>>(ei, zl, sums2, E);
    // 6) final combine
    final_kernel<<<(N + B - 1) / B, B, 0, stream>>>(sums2, counts, zr, outf, N);

    (void)n_in; (void)out_size; (void)ws_size;
}